// ChamferLoss_66022237274636
// MI455X (gfx1250) — compile-verified
//
#include <hip/hip_runtime.h>
#include <stdint.h>

typedef float v2f __attribute__((ext_vector_type(2)));
typedef float v8f __attribute__((ext_vector_type(8)));

constexpr int Bsz  = 8;
constexpr int Dim  = 4096;                       // M == N == 4096
constexpr int WAVES_PER_WG = 8;                  // wave32 -> 256 threads
constexpr int ROWS_PER_WG  = WAVES_PER_WG * 16;  // 128 rows per workgroup
constexpr int THREADS      = WAVES_PER_WG * 32;  // 256

__device__ __forceinline__ unsigned long long pack_du(float d2, unsigned idx) {
    d2 = d2 > 0.0f ? d2 : 0.0f;   // cndmask clamp: keeps uint ordering == float ordering
    return (((unsigned long long)__float_as_uint(d2)) << 32) | (unsigned long long)idx;
}

// ---------------- fused distance + row argmin (one direction) ----------------
// For each "row" point p in P[b,:,m]: find min_n ||p - Q[b,:,n]||^2 and argmin.
// Called twice with P/Q swapped to get both Chamfer directions.
__global__ void __launch_bounds__(THREADS)
chamfer_rowmin_kernel(const float* __restrict__ P,   // [B,3,Dim] row points
                      const float* __restrict__ Q,   // [B,3,Dim] column points
                      unsigned long long* __restrict__ outPacked)  // [B*Dim]
{
    __shared__ float colS[3 * Dim];               // 48 KB staged column points (x|y|z)

    const int blocksPerBatch = Dim / ROWS_PER_WG;
    const int b    = blockIdx.x / blocksPerBatch;
    const int mblk = blockIdx.x % blocksPerBatch;
    const int tid  = threadIdx.x;
    const int wave = tid >> 5;
    const int lane = tid & 31;
    const int half = lane >> 4;      // 0: lanes 0-15, 1: lanes 16-31
    const int l16  = lane & 15;
    const int m_base = mblk * ROWS_PER_WG + wave * 16;

    // ---- async-stage this batch's column points into LDS (gfx1250 ASYNCcnt path) ----
    {
        const float* gsrc = Q + (size_t)b * 3 * Dim;         // 48 KB contiguous
        unsigned ldsBase = (unsigned)(uintptr_t)&colS[0];    // LDS offset = low 32 bits
        for (int k = tid; k < (3 * Dim) / 4; k += THREADS) {
            unsigned loff = ldsBase + k * 16;
            const float* g = gsrc + k * 4;
            asm volatile("global_load_async_to_lds_b128 %0, %1, off"
                         :: "v"(loff), "v"(g) : "memory");
        }
        asm volatile("s_wait_asynccnt 0" ::: "memory");
    }
    __syncthreads();

    const float* rowX = P + (size_t)b * 3 * Dim;
    const float* rowY = rowX + Dim;
    const float* rowZ = rowY + Dim;

    // A operand (16x4 f32, fixed per wave): lanes 0-15 -> K=0,1 ; lanes 16-31 -> K=2,3
    const int mA = m_base + l16;
    float ax = rowX[mA], ay = rowY[mA], az = rowZ[mA];
    v2f a;
    a[0] = half ? (-2.0f * az) : (-2.0f * ax);
    a[1] = half ? 1.0f         : (-2.0f * ay);

    // C operand: ||p_m||^2 broadcast along each row of the 16x16 tile
    const int mrow0 = m_base + half * 8;
    v8f c;
    #pragma unroll
    for (int r = 0; r < 8; ++r) {
        int m = mrow0 + r;
        float x = rowX[m], y = rowY[m], z = rowZ[m];
        c[r] = x * x + y * y + z * z;
    }

    float    rowD[8];
    unsigned rowI[8];
    #pragma unroll
    for (int r = 0; r < 8; ++r) { rowD[r] = 3.4e38f; rowI[r] = 0u; }

    for (int t = 0; t < Dim / 16; ++t) {
        const int n = t * 16 + l16;
        float qx = colS[n], qy = colS[Dim + n], qz = colS[2 * Dim + n];
        float nrm = qx * qx + qy * qy + qz * qz;
        // B operand (4x16 f32): lanes 0-15 -> K=0,1 ; lanes 16-31 -> K=2,3 (branchless)
        v2f bv;
        bv[0] = half ? qz  : qx;
        bv[1] = half ? nrm : qy;

        // D[m,n] = ||p||^2 - 2 p.q + ||q||^2  (one 16x16 tile per instruction)
        v8f d = __builtin_amdgcn_wmma_f32_16x16x4_f32(
            false, a, false, bv, (short)0, c, false, false);

        // independent running argmin per row (ascending n => strict < keeps first index)
        #pragma unroll
        for (int r = 0; r < 8; ++r) {
            bool lt = d[r] < rowD[r];
            rowD[r] = lt ? d[r] : rowD[r];
            rowI[r] = lt ? (unsigned)n : rowI[r];
        }
    }

    // cross-lane reduction within each 16-lane half (rows r and r+8 simultaneously);
    // packed u64 compare gives min-distance with lowest-index tie-break.
    #pragma unroll
    for (int r = 0; r < 8; ++r) {
        unsigned long long v = pack_du(rowD[r], rowI[r]);
        #pragma unroll
        for (int mask = 1; mask <= 8; mask <<= 1) {
            unsigned long long o = __shfl_xor(v, mask, 32);
            if (o < v) v = o;
        }
        if (l16 == 0) outPacked[(size_t)b * Dim + mrow0 + r] = v;
    }
}

// ---------------- final deterministic reduction to scalar ----------------
__global__ void __launch_bounds__(1024)
chamfer_reduce_kernel(const unsigned long long* __restrict__ fwdPacked,  // [B*Dim] src->dst
                      const unsigned long long* __restrict__ bwdPacked,  // [B*Dim] dst->src
                      const float* __restrict__ sigma_src,   // [B,Dim]
                      const float* __restrict__ sigma_dst,   // [B,Dim]
                      float* __restrict__ out)
{
    __shared__ float red[1024];
    const int total = Bsz * Dim;
    const float invTotal = 1.0f / (float)total;
    float acc = 0.0f;

    for (int i = threadIdx.x; i < total; i += 1024) {
        int b = i / Dim;
        // forward: row = src point, index = dst point
        unsigned long long p = fwdPacked[i];
        float dist = sqrtf(fmaxf(__uint_as_float((uint32_t)(p >> 32)), 0.0f));
        int idx = (int)(uint32_t)p;
        float s = 0.5f * (sigma_src[i] + sigma_dst[(size_t)b * Dim + idx]);
        acc += dist * s * invTotal;
        // backward: row = dst point, index = src point
        unsigned long long q = bwdPacked[i];
        float distb = sqrtf(fmaxf(__uint_as_float((uint32_t)(q >> 32)), 0.0f));
        int idxb = (int)(uint32_t)q;
        float sb = 0.5f * (sigma_dst[i] + sigma_src[(size_t)b * Dim + idxb]);
        acc += distb * sb * invTotal;
    }

    red[threadIdx.x] = acc;
    __syncthreads();
    for (int s = 512; s > 0; s >>= 1) {
        if (threadIdx.x < (unsigned)s) red[threadIdx.x] += red[threadIdx.x + s];
        __syncthreads();
    }
    if (threadIdx.x == 0) out[0] = red[0];
}

extern "C" void kernel_launch(void* const* d_in, const int* in_sizes, int n_in,
                              void* d_out, int out_size, void* d_ws, size_t ws_size,
                              hipStream_t stream) {
    (void)in_sizes; (void)n_in; (void)out_size; (void)ws_size;
    const float* pc_src    = (const float*)d_in[0];
    const float* pc_dst    = (const float*)d_in[1];
    const float* sigma_src = (const float*)d_in[2];
    const float* sigma_dst = (const float*)d_in[3];

    unsigned long long* fwdPacked = (unsigned long long*)d_ws;            // 256 KB
    unsigned long long* bwdPacked = fwdPacked + (size_t)Bsz * Dim;        // 256 KB

    const int grid = Bsz * (Dim / ROWS_PER_WG);   // 256 workgroups per direction

    // src -> dst
    chamfer_rowmin_kernel<<<grid, THREADS, 0, stream>>>(pc_src, pc_dst, fwdPacked);
    // dst -> src (same kernel, roles swapped)
    chamfer_rowmin_kernel<<<grid, THREADS, 0, stream>>>(pc_dst, pc_src, bwdPacked);

    chamfer_reduce_kernel<<<1, 1024, 0, stream>>>(
        fwdPacked, bwdPacked, sigma_src, sigma_dst, (float*)d_out);
}